// WTKan2d_72696616452736
// MI455X (gfx1250) — compile-verified
//
#include <hip/hip_runtime.h>

typedef __attribute__((ext_vector_type(2))) float    v2f;
typedef __attribute__((ext_vector_type(8))) float    v8f;
typedef __attribute__((ext_vector_type(4))) unsigned v4u;
typedef __attribute__((ext_vector_type(8))) int      v8i;
typedef __attribute__((ext_vector_type(4))) int      v4i;

#define NB 4
#define NC 32
#define NH 256
#define NW 256
#define BCS 66          // bases-chunk row stride in floats (bank-conflict padding)
#define N_SPL_PAIRS (256 * 128 * 2)   // spline B-pack: 256 k-tiles x 128 cols x 2 halves
#define N_BASE_PAIRS (32 * 128 * 2)   // base   B-pack:  32 k-tiles x 128 cols x 2 halves

__device__ __forceinline__ float siluf(float v) {
    return v / (1.0f + __expf(-v));
}

// 8 cubic B-spline bases on uniform grid g[m] = (m-3)*0.4 - 1, m = 0..11
// (exactly the reference Cox-de Boor recursion, SPLINE_ORDER=3, GRID_SIZE=5)
__device__ __forceinline__ void bspline8(float t, float* __restrict__ out) {
    const float h = 0.4f;
    float b0[11];
#pragma unroll
    for (int m = 0; m < 11; ++m) {
        float gl = -1.0f + (float)(m - 3) * h;
        b0[m] = (t >= gl && t < gl + h) ? 1.0f : 0.0f;
    }
    float b1[10];
#pragma unroll
    for (int m = 0; m < 10; ++m) {
        float gm = -1.0f + (float)(m - 3) * h;
        b1[m] = ((t - gm) * b0[m] + (gm + 2.0f * h - t) * b0[m + 1]) * 2.5f;       // 1/h
    }
    float b2[9];
#pragma unroll
    for (int m = 0; m < 9; ++m) {
        float gm = -1.0f + (float)(m - 3) * h;
        b2[m] = ((t - gm) * b1[m] + (gm + 3.0f * h - t) * b1[m + 1]) * 1.25f;      // 1/(2h)
    }
#pragma unroll
    for (int m = 0; m < 8; ++m) {
        float gm = -1.0f + (float)(m - 3) * h;
        out[m] = ((t - gm) * b2[m] + (gm + 4.0f * h - t) * b2[m + 1]) * 0.8333333333f; // 1/(3h)
    }
}

// Haar analysis value for band k at (il, j) from the LDS x tile (32 x 256)
__device__ __forceinline__ float wt_at(const float* __restrict__ xs, int k, int il, int j) {
    float x00 = xs[(2 * il) * NW + 2 * j];
    float x01 = xs[(2 * il) * NW + 2 * j + 1];
    float x10 = xs[(2 * il + 1) * NW + 2 * j];
    float x11 = xs[(2 * il + 1) * NW + 2 * j + 1];
    float r;
    if (k == 0)      r = x00 + x01 + x10 + x11;
    else if (k == 1) r = x00 + x01 - x10 - x11;
    else if (k == 2) r = x00 - x01 + x10 - x11;
    else             r = x00 - x01 - x10 + x11;
    return 0.5f * r;
}

// ---------------------------------------------------------------------------
// Prep: pack B matrices into WMMA-fragment-ready pair layout.
// pair idx P = (kt*128 + col)*2 + s2  holds { Bmat[K0][col], Bmat[K0+1][col] },
// K0 = kt*4 + s2*2.  Spline: Bmat[K][o] = spline_w[o][K/8][K%8] * scaler[o][K/8].
// Base:   Bmat[K][o] = base_w[o][K].
// ---------------------------------------------------------------------------
__global__ __launch_bounds__(256) void wtkan_prep(
    const float* __restrict__ base_w,    // (128,128) row-major (o,i)
    const float* __restrict__ spline_w,  // (128,128,8) row-major (o,i,g)
    const float* __restrict__ scaler,    // (128,128) (o,i)
    float* __restrict__ BpkSpl,          // N_SPL_PAIRS pairs (float2 each)
    float* __restrict__ BpkBase)         // N_BASE_PAIRS pairs
{
    int idx = blockIdx.x * blockDim.x + threadIdx.x;
    if (idx < N_SPL_PAIRS) {
        int s2  = idx & 1;
        int col = (idx >> 1) & 127;
        int kt  = idx >> 8;
        int K0  = kt * 4 + s2 * 2;
        int j = K0 >> 3, g = K0 & 7;     // K0 even => K0,K0+1 share the same j
        float sc = scaler[col * 128 + j];
        BpkSpl[idx * 2 + 0] = spline_w[(col * 128 + j) * 8 + g]     * sc;
        BpkSpl[idx * 2 + 1] = spline_w[(col * 128 + j) * 8 + g + 1] * sc;
    } else if (idx < N_SPL_PAIRS + N_BASE_PAIRS) {
        int p  = idx - N_SPL_PAIRS;
        int s2 = p & 1;
        int col = (p >> 1) & 127;
        int kt  = p >> 8;
        int K0  = kt * 4 + s2 * 2;
        BpkBase[p * 2 + 0] = base_w[col * 128 + K0];
        BpkBase[p * 2 + 1] = base_w[col * 128 + K0 + 1];
    }
}

// ---------------------------------------------------------------------------
// Fused main kernel: TDM x-tile stage -> Haar analysis -> KAN (spline + base
// GEMMs via V_WMMA_F32_16X16X4_F32) -> Haar synthesis + depthwise 5x5 skip.
// One workgroup per (b, c, 16-half-row block). 256 threads = 8 waves.
// Wave w: wavelet band k = w&3 (16 GEMM rows), output cols 64*(w>>2)..+63.
// ---------------------------------------------------------------------------
__global__ __launch_bounds__(256) void wtkan_main(
    const float* __restrict__ x,         // (B,C,256,256)
    const float* __restrict__ conv_w,    // (C,1,5,5)
    const float* __restrict__ conv_b,    // (C)
    const float* __restrict__ BpkSpl,
    const float* __restrict__ BpkBase,
    float* __restrict__ out)             // (B,C,256,256)
{
    __shared__ float xs[32 * NW];   // 32 KB: full-res x tile (32 rows x 256)
    __shared__ float un[8192];      // 32 KB union: bases chunk (64 x BCS) / z buffer (64 x 128)

    const int t  = threadIdx.x;
    const int bx = blockIdx.x;
    const int ib = bx & 7;
    const int c  = (bx >> 3) & 31;
    const int b  = bx >> 8;
    const int i0 = ib * 16;     // half-res row block start
    const int r0 = 2 * i0;      // full-res row start (32 rows)

    const float* __restrict__ xin = x + (size_t)(b * NC + c) * NH * NW;

    const int lane = t & 31;
    const int wv   = t >> 5;

    // ---- stage x tile (contiguous 8192 floats) into LDS via Tensor Data Mover ----
    // 1-D descriptor: data_size=4B, tensor_dim0 = tile_dim0 = 8192, no pad/iterate,
    // workgroup_mask = 0 (not dispatched in a cluster). Wave 0 issues the DMA and
    // drains its TENSORcnt; the workgroup barrier publishes the tile.
    if (wv == 0) {
        unsigned lds_addr = (unsigned)(size_t)(void*)xs;          // flat addr[31:0] == LDS offset
        unsigned long long ga = (unsigned long long)(size_t)(xin + (size_t)r0 * NW);
        v4u g0;
        g0[0] = 1u;                                               // count=1, user D#
        g0[1] = lds_addr;                                         // lds_addr
        g0[2] = (unsigned)(ga & 0xffffffffu);                     // global_addr[31:0]
        g0[3] = (unsigned)((ga >> 32) & 0x01ffffffu) | (2u << 30);// global_addr[56:32] | type=2
        v8i g1;
        g1[0] = 0x20000;             // data_size = 2 (4 bytes); mask/pad/iterate = 0
        g1[1] = (int)(8192u << 16);  // tensor_dim0[15:0] in [63:48]  (= 8192)
        g1[2] = (int)(1u << 16);     // tensor_dim0[31:16]=0, tensor_dim1[15:0]=1
        g1[3] = (int)(8192u << 16);  // tensor_dim1[31:16]=0, tile_dim0 = 8192
        g1[4] = 0;                   // tile_dim1 = 0 (unused), tile_dim2 = 0
        g1[5] = 8192;                // tensor_dim0_stride[31:0]
        g1[6] = 0;
        g1[7] = 0;
        v4i gz4 = {0, 0, 0, 0};
        v8i gz8 = {0, 0, 0, 0, 0, 0, 0, 0};
        __builtin_amdgcn_tensor_load_to_lds(g0, g1, gz4, gz4, gz8, 0);
        __builtin_amdgcn_s_wait_tensorcnt(0);
    }
    __syncthreads();

    const int ksub = wv & 3;        // wavelet band handled by this wave
    const int colbase = (wv >> 2) * 64;
    const int lm = lane & 15;
    const int s2 = lane >> 4;       // half-wave selector (K offset 0 / 2)

    v8f acc[4] = {};                // 4 x 16x16 fp32 output tiles per wave

    // ================= spline GEMM: K = 1024 in 16 chunks of 64 =================
    for (int jc = 0; jc < 16; ++jc) {
        // fill bases chunk: rows 0..63 (= band*16 + il), Klocal = jj*8 + g
#pragma unroll
        for (int u = 0; u < 2; ++u) {
            int pp  = u * 256 + t;        // 512 (row, jj) pairs
            int row = pp >> 3;
            int jj  = pp & 7;
            float tv = wt_at(xs, row >> 4, row & 15, jc * 8 + jj);
            float bs[8];
            bspline8(tv, bs);
            float* d = &un[row * BCS + jj * 8];
#pragma unroll
            for (int g = 0; g < 8; ++g) d[g] = bs[g];
        }
        __syncthreads();

        for (int ktl = 0; ktl < 16; ++ktl) {
            int kg = jc * 16 + ktl;       // global k-tile (K base = 4*kg)
            v2f a = *(const v2f*)&un[(ksub * 16 + lm) * BCS + ktl * 4 + s2 * 2];
#pragma unroll
            for (int tt = 0; tt < 4; ++tt) {
                int col = colbase + tt * 16 + lm;
                v2f bfr = *(const v2f*)(BpkSpl + (((size_t)kg * 128 + col) * 2 + s2) * 2);
                acc[tt] = __builtin_amdgcn_wmma_f32_16x16x4_f32(
                    false, a, false, bfr, (short)0, acc[tt], false, false);
            }
        }
        __syncthreads();
    }

    // ================= base GEMM: K = 128 in 2 chunks of 64 (A = silu(wt)) =====
    for (int fc = 0; fc < 2; ++fc) {
#pragma unroll
        for (int u = 0; u < 16; ++u) {
            int pp  = u * 256 + t;        // 4096 (row, f) pairs
            int row = pp >> 6;
            int f   = pp & 63;
            un[row * BCS + f] = siluf(wt_at(xs, row >> 4, row & 15, fc * 64 + f));
        }
        __syncthreads();

        for (int ktl = 0; ktl < 16; ++ktl) {
            int kg = fc * 16 + ktl;
            v2f a = *(const v2f*)&un[(ksub * 16 + lm) * BCS + ktl * 4 + s2 * 2];
#pragma unroll
            for (int tt = 0; tt < 4; ++tt) {
                int col = colbase + tt * 16 + lm;
                v2f bfr = *(const v2f*)(BpkBase + (((size_t)kg * 128 + col) * 2 + s2) * 2);
                acc[tt] = __builtin_amdgcn_wmma_f32_16x16x4_f32(
                    false, a, false, bfr, (short)0, acc[tt], false, false);
            }
        }
        __syncthreads();
    }

    // ---- spill z tiles to LDS as zbuf[k][il][j] (stride 128) ----
#pragma unroll
    for (int tt = 0; tt < 4; ++tt) {
        int col = colbase + tt * 16 + lm;
#pragma unroll
        for (int v = 0; v < 8; ++v) {
            int M = v + s2 * 8;                        // C/D layout: VGPR v -> M=v / v+8
            un[(ksub * 16 + M) * 128 + col] = acc[tt][v];
        }
    }
    __syncthreads();

    // ================= Haar synthesis + depthwise 5x5 skip + store ============
    float wreg[25];
#pragma unroll
    for (int q = 0; q < 25; ++q) wreg[q] = conv_w[c * 25 + q];
    const float bias = conv_b[c];
    float* __restrict__ outp = out + (size_t)(b * NC + c) * NH * NW;

    for (int m = 0; m < 32; ++m) {
        const int r = m, cc = t;
        const int il = r >> 1, p = r & 1, j = cc >> 1, q = cc & 1;
        float z0 = un[(0 * 16 + il) * 128 + j];
        float z1 = un[(1 * 16 + il) * 128 + j];
        float z2 = un[(2 * 16 + il) * 128 + j];
        float z3 = un[(3 * 16 + il) * 128 + j];
        float s1 = p ? -1.0f : 1.0f;
        float sq = q ? -1.0f : 1.0f;
        float y = 0.5f * (z0 + s1 * z1 + sq * z2 + s1 * sq * z3);

        float acv = bias;
#pragma unroll
        for (int dy = -2; dy <= 2; ++dy) {
            int lr = r + dy;
            int gr = r0 + r + dy;
            bool inRow = (gr >= 0 && gr < NH);
#pragma unroll
            for (int dx = -2; dx <= 2; ++dx) {
                int gc = cc + dx;
                float xv = 0.0f;
                if (inRow && gc >= 0 && gc < NW) {
                    xv = (lr >= 0 && lr < 32) ? xs[lr * NW + gc]
                                              : xin[(size_t)gr * NW + gc];
                }
                acv = fmaf(wreg[(dy + 2) * 5 + (dx + 2)], xv, acv);
            }
        }
        outp[(size_t)(r0 + r) * NW + cc] = y + acv;
    }
}

// ---------------------------------------------------------------------------
extern "C" void kernel_launch(void* const* d_in, const int* in_sizes, int n_in,
                              void* d_out, int out_size, void* d_ws, size_t ws_size,
                              hipStream_t stream) {
    const float* x        = (const float*)d_in[0];
    const float* conv_w   = (const float*)d_in[1];
    const float* conv_b   = (const float*)d_in[2];
    const float* base_w   = (const float*)d_in[3];
    const float* spline_w = (const float*)d_in[4];
    const float* scaler   = (const float*)d_in[5];
    float* out = (float*)d_out;

    float* BpkSpl  = (float*)d_ws;                                        // 512 KB
    float* BpkBase = (float*)((char*)d_ws + (size_t)N_SPL_PAIRS * 2 * 4); // 64 KB

    wtkan_prep<<<(N_SPL_PAIRS + N_BASE_PAIRS + 255) / 256, 256, 0, stream>>>(
        base_w, spline_w, scaler, BpkSpl, BpkBase);
    wtkan_main<<<NB * NC * 8, 256, 0, stream>>>(
        x, conv_w, conv_b, BpkSpl, BpkBase, out);
}